// FNsGCN_712964571463
// MI455X (gfx1250) — compile-verified
//
#include <hip/hip_runtime.h>
#include <hip/hip_bf16.h>

typedef __attribute__((ext_vector_type(2))) float v2f;
typedef __attribute__((ext_vector_type(8))) float v8f;

// ---------------------------------------------------------------------------
// Degree: deg[dst] += 1 per edge (computed once, reused by both layers)
// ---------------------------------------------------------------------------
__global__ void degree_kernel(const long long* __restrict__ dst, float* __restrict__ deg,
                              long long E) {
    long long e = (long long)blockIdx.x * blockDim.x + threadIdx.x;
    if (e < E) {
        atomicAdd(&deg[dst[e]], 1.0f);
    }
}

// ---------------------------------------------------------------------------
// Edge scatter: agg[dst] += x[src], 32 lanes per edge, float4 per lane.
// Working set (x + agg ~ 100MB) fits in the 192MB L2, so these atomics
// resolve at L2 rates.
// ---------------------------------------------------------------------------
__global__ void scatter_add_kernel(const float* __restrict__ x,
                                   const long long* __restrict__ src,
                                   const long long* __restrict__ dst,
                                   float* __restrict__ agg, long long E) {
    long long idx = (long long)blockIdx.x * blockDim.x + threadIdx.x;
    long long total = E * 32;
    if (idx >= total) return;
    long long e = idx >> 5;
    int c = (int)(idx & 31);
    long long s = src[e];
    long long d = dst[e];
    const float4 v = *(const float4*)(x + s * 128 + c * 4);
    float* o = agg + d * 128 + c * 4;
    atomicAdd(o + 0, v.x);
    atomicAdd(o + 1, v.y);
    atomicAdd(o + 2, v.z);
    atomicAdd(o + 3, v.w);
}

// ---------------------------------------------------------------------------
// Fused SAGE layer GEMM using V_WMMA_F32_16X16X4_F32 (exact fp32 path).
//   out[i][j] = act( (agg[i]/max(deg[i],1)) @ Wl + x[i] @ Wr + bias[j] )
//
// One wave (blockDim == 32) computes a full 16-row x (CT*16)-col output slab:
//   - A fragments (mean and x rows) are loaded ONCE per K-step and reused
//     across all CT column tiles (8x less A traffic than tile-per-wave).
//   - CT independent accumulators give the WMMA pipe ILP instead of one
//     serial RAW chain.
//   - In-place out == agg is race-free: this wave reads all 16 of its A rows
//     through the whole K loop before any store, and no other wave touches
//     those rows.
//
// Fragment layouts per CDNA5 ISA 7.12.2:
//   A 16x4 f32:  2 VGPRs; lane%16 = M-row; lane/16 selects K pair {0,1}/{2,3}
//   B 4x16 f32:  2 VGPRs; mirrored (K-row striped across lanes, N = lane%16)
//   C/D 16x16:   8 VGPRs; VGPR v -> M = v + 8*(lane/16), N = lane%16
// ---------------------------------------------------------------------------
template <bool HAS_MEAN, bool RELU, int CT>
__global__ void __launch_bounds__(32)
sage_gemm_kernel(const float* __restrict__ agg,
                 const float* __restrict__ deg,
                 const float* __restrict__ xr,
                 const float* __restrict__ Wl,
                 const float* __restrict__ Wr,
                 const float* __restrict__ bias,
                 float* __restrict__ out,
                 int Nrows) {
    constexpr int H = CT * 16;

    const int lane = threadIdx.x & 31;
    const int l16  = lane & 15;
    const int half = lane >> 4;
    const int row0 = blockIdx.x * 16;

    // Clamp row for loads so EXEC stays all-ones through the WMMAs.
    const int rowL  = row0 + l16;
    const int rowLd = (rowL < Nrows) ? rowL : (Nrows - 1);

    v8f acc[CT];
#pragma unroll
    for (int ct = 0; ct < CT; ++ct) {
        const float bv = bias[ct * 16 + l16];
#pragma unroll
        for (int v = 0; v < 8; ++v) acc[ct][v] = bv;
    }

    float rdeg = 0.0f;
    const float* arow_mean = nullptr;
    if (HAS_MEAN) {
        const float dg = deg[rowLd];
        rdeg = 1.0f / fmaxf(dg, 1.0f);
        arow_mean = agg + (size_t)rowLd * 128;
    }
    const float* arow_x = xr + (size_t)rowLd * 128;

#pragma unroll 4
    for (int kk = 0; kk < 128; kk += 4) {
        const int ka = kk + half * 2;   // this lane's K pair base
        if (HAS_MEAN) {
            const float2 am = *(const float2*)(arow_mean + ka);
            v2f av;  av.x = am.x * rdeg;  av.y = am.y * rdeg;
            const float* w0 = Wl + (size_t)(ka + 0) * H;
            const float* w1 = Wl + (size_t)(ka + 1) * H;
#pragma unroll
            for (int ct = 0; ct < CT; ++ct) {
                v2f bv;
                bv.x = w0[ct * 16 + l16];
                bv.y = w1[ct * 16 + l16];
                acc[ct] = __builtin_amdgcn_wmma_f32_16x16x4_f32(
                    false, av, false, bv, (short)0, acc[ct], false, false);
            }
        }
        {
            const float2 ax = *(const float2*)(arow_x + ka);
            v2f av;  av.x = ax.x;  av.y = ax.y;
            const float* w0 = Wr + (size_t)(ka + 0) * H;
            const float* w1 = Wr + (size_t)(ka + 1) * H;
#pragma unroll
            for (int ct = 0; ct < CT; ++ct) {
                v2f bv;
                bv.x = w0[ct * 16 + l16];
                bv.y = w1[ct * 16 + l16];
                acc[ct] = __builtin_amdgcn_wmma_f32_16x16x4_f32(
                    false, av, false, bv, (short)0, acc[ct], false, false);
            }
        }
    }

#pragma unroll
    for (int ct = 0; ct < CT; ++ct) {
#pragma unroll
        for (int v = 0; v < 8; ++v) {
            float val = acc[ct][v];
            if (RELU) val = fmaxf(val, 0.0f);
            const int row = row0 + v + half * 8;
            if (row < Nrows) {
                out[(size_t)row * H + ct * 16 + l16] = val;
            }
        }
    }
}

// ---------------------------------------------------------------------------
// Host-side launch
// ---------------------------------------------------------------------------
extern "C" void kernel_launch(void* const* d_in, const int* in_sizes, int n_in,
                              void* d_out, int out_size, void* d_ws, size_t ws_size,
                              hipStream_t stream) {
    const int D = 128;

    const long long Nn = in_sizes[0] / D;      // 100000
    const long long E  = in_sizes[1] / 2;      // 1600000

    const float*     x    = (const float*)d_in[0];
    const long long* ei   = (const long long*)d_in[1];   // int64 in reference
    const float*     W1l  = (const float*)d_in[2];
    const float*     b1   = (const float*)d_in[3];
    const float*     W1r  = (const float*)d_in[4];
    const float*     W2l  = (const float*)d_in[5];
    const float*     b2   = (const float*)d_in[6];
    const float*     W2r  = (const float*)d_in[7];
    const float*     Wout = (const float*)d_in[8];
    const float*     bout = (const float*)d_in[9];
    float*           out  = (float*)d_out;

    const long long* src = ei;
    const long long* dst = ei + E;

    // Workspace layout (1MB-aligned regions):
    //   deg : N floats
    //   agg : N*128 floats (reused for layer-2 aggregate, then holds h2)
    //   h1  : N*128 floats
    char* ws = (char*)d_ws;
    const size_t MB = (size_t)1 << 20;
    float* deg = (float*)ws;
    float* agg = (float*)(ws + MB);
    float* h1  = (float*)(ws + MB + ((size_t)Nn * D * sizeof(float) + MB - 1) / MB * MB);

    const size_t aggBytes = (size_t)Nn * D * sizeof(float);

    const int TPB = 256;
    const long long scatterThreads = E * 32;
    const int scatterBlocks = (int)((scatterThreads + TPB - 1) / TPB);
    const int degBlocks = (int)((E + TPB - 1) / TPB);
    const int rowTiles = (int)((Nn + 15) / 16);

    // ---- degrees (once) ----
    hipMemsetAsync(deg, 0, (size_t)Nn * sizeof(float), stream);
    degree_kernel<<<degBlocks, TPB, 0, stream>>>(dst, deg, E);

    // ---- layer 1: aggregate x, then fused GEMM + ReLU -> h1 ----
    hipMemsetAsync(agg, 0, aggBytes, stream);
    scatter_add_kernel<<<scatterBlocks, TPB, 0, stream>>>(x, src, dst, agg, E);
    sage_gemm_kernel<true, true, 8><<<rowTiles, 32, 0, stream>>>(
        agg, deg, x, W1l, W1r, b1, h1, (int)Nn);

    // ---- layer 2: aggregate h1, fused GEMM + ReLU, in place over agg ----
    hipMemsetAsync(agg, 0, aggBytes, stream);
    scatter_add_kernel<<<scatterBlocks, TPB, 0, stream>>>(h1, src, dst, agg, E);
    sage_gemm_kernel<true, true, 8><<<rowTiles, 32, 0, stream>>>(
        agg, deg, h1, W2l, W2r, b2, agg /* in-place -> h2 */, (int)Nn);

    // ---- output head: out = h2 @ Wout + bout (no mean branch, no relu) ----
    sage_gemm_kernel<false, false, 4><<<rowTiles, 32, 0, stream>>>(
        nullptr, nullptr, agg, nullptr, Wout, bout, out, (int)Nn);
}